// RFA_SPDConv_Fourier_SDI_17678085390725
// MI455X (gfx1250) — compile-verified
//
#include <hip/hip_runtime.h>
#include <math.h>

typedef __attribute__((ext_vector_type(2))) float v2f;
typedef __attribute__((ext_vector_type(8))) float v8f;

#define WMMA_F32X4(a, b, c) \
    __builtin_amdgcn_wmma_f32_16x16x4_f32(false, (a), false, (b), (short)0, (c), false, false)

// Packed WMMA-B layout index: element (k,n) of a KxN matrix (N=256) lives at
//   (((k>>2)*256 + n)*2 + ((k>>1)&1))*2 + (k&1)
// so a lane's B operand pair (B[k,n], B[k+1,n]) is one aligned 8-byte load,
// and advancing k by 4 advances the address by a constant 1024 floats.
__device__ __forceinline__ size_t bpack_idx(int k, int n) {
    return ((size_t)(((k >> 2) * 256 + n) * 2 + ((k >> 1) & 1)) << 1) | (k & 1);
}

// ---------------------------------------------------------------------------
// Kernel 0: twiddle matrices.
//   Natural:  twc[m*512+k] = cos(2*pi*m*k/512), tws = sin(...)   (A-side, col pass)
//   Packed :  twpc/twps = cols 0..255 in WMMA-B pair layout       (B-side, row pass)
// ---------------------------------------------------------------------------
__global__ void k_twiddle(float* __restrict__ twc, float* __restrict__ tws,
                          float* __restrict__ twpc, float* __restrict__ twps) {
    int idx = blockIdx.x * blockDim.x + threadIdx.x;   // 0..262143
    int i = idx >> 9;                                  // k-row
    int j = idx & 511;                                 // n-col
    int t = (i * j) & 511;                             // exp period 512
    float th = (float)t * (6.28318530717958647692f / 512.0f);
    float c = cosf(th), s = sinf(th);
    twc[idx] = c;
    tws[idx] = s;
    if (j < 256) {
        size_t p = bpack_idx(i, j);
        twpc[p] = c;
        twps[p] = s;
    }
}

// ---------------------------------------------------------------------------
// Kernel 1: space-to-depth 1x1 conv + BN + SiLU, zero-padded to 512x512
// x: (16,1,1022,1022) -> ypad: (16,512,512)
// ---------------------------------------------------------------------------
__global__ void k_spd_pad(const float* __restrict__ x,
                          const float* __restrict__ w4,
                          const float* __restrict__ g,
                          const float* __restrict__ b,
                          float* __restrict__ ypad) {
    int idx = blockIdx.x * blockDim.x + threadIdx.x;   // 16*512*512
    int j  = idx & 511;
    int i  = (idx >> 9) & 511;
    int bb = idx >> 18;
    float v = 0.0f;
    if (i < 511 && j < 511) {
        const float* xb = x + (size_t)bb * (1022u * 1022u);
        int r = 2 * i, c = 2 * j;
        float c0 = xb[(size_t)r * 1022 + c];           // x[::2,  ::2]
        float c1 = xb[(size_t)(r + 1) * 1022 + c];     // x[1::2, ::2]
        float c2 = xb[(size_t)r * 1022 + c + 1];       // x[::2, 1::2]
        float c3 = xb[(size_t)(r + 1) * 1022 + c + 1]; // x[1::2,1::2]
        float s = c0 * w4[0] + c1 * w4[1] + c2 * w4[2] + c3 * w4[3];
        float scale = g[0] * rsqrtf(1.0f + 1e-5f);
        s = s * scale + b[0];
        v = s / (1.0f + __expf(-s));                   // SiLU
    }
    ypad[idx] = v;
}

// ---------------------------------------------------------------------------
// Kernel 2: row rfft as DFT matmul, two images per wave (twiddle B reuse).
//   Yr =  Y @ cos,  Yi = -(Y @ sin), cols 0..255 only.
// Results stored in WMMA-B pair layout for the column pass.
// Per k-step: 4 b64 loads (all immediate-offset) -> 4 WMMA.
// ---------------------------------------------------------------------------
__global__ void __launch_bounds__(256)
k_rowdft(const float* __restrict__ ypad,
         const float* __restrict__ twpc, const float* __restrict__ twps,
         float* __restrict__ Yrp, float* __restrict__ Yip) {
    int wave = blockIdx.x * 8 + (threadIdx.x >> 5);    // 4096 waves total
    int lane = threadIdx.x & 31;
    int pair = wave >> 9;                              // image pair 0..7
    int rem  = wave & 511;                             // 32 x 16 tiles
    int m0 = (rem >> 4) << 4;
    int n0 = (rem & 15) << 4;
    int ln    = lane & 15;
    int khalf = lane >> 4;                             // 0 or 1
    int koff  = khalf << 1;                            // 0 or 2
    int arow  = m0 + ln;

    // Hoisted bases; per k-step strides are compile-time constants:
    //   A: +4 floats, packed B: +1024 floats.
    const float* pa0 = ypad + (size_t)(2 * pair)     * (512u * 512u)
                            + (size_t)arow * 512 + koff;
    const float* pa1 = ypad + (size_t)(2 * pair + 1) * (512u * 512u)
                            + (size_t)arow * 512 + koff;
    size_t bp0 = ((size_t)((n0 + ln) * 2 + khalf)) << 1;
    const float* pbc = twpc + bp0;
    const float* pbs = twps + bp0;

    v8f accr0 = {}, acci0 = {}, accr1 = {}, acci1 = {};
#pragma unroll 8
    for (int kb = 0; kb < 128; kb++) {
        v2f a0 = *(const v2f*)(pa0 + (size_t)kb * 4);
        v2f a1 = *(const v2f*)(pa1 + (size_t)kb * 4);
        v2f bc = *(const v2f*)(pbc + (size_t)kb * 1024);
        v2f bs = *(const v2f*)(pbs + (size_t)kb * 1024);
        accr0 = WMMA_F32X4(a0, bc, accr0);
        acci0 = WMMA_F32X4(a0, bs, acci0);
        accr1 = WMMA_F32X4(a1, bc, accr1);
        acci1 = WMMA_F32X4(a1, bs, acci1);
    }
    float* yr0 = Yrp + (size_t)(2 * pair)     * (512u * 256u);
    float* yi0 = Yip + (size_t)(2 * pair)     * (512u * 256u);
    float* yr1 = Yrp + (size_t)(2 * pair + 1) * (512u * 256u);
    float* yi1 = Yip + (size_t)(2 * pair + 1) * (512u * 256u);
    int rbase = m0 + (khalf << 3);                     // +8 for lanes 16..31
    int col = n0 + ln;
#pragma unroll
    for (int v = 0; v < 8; v++) {
        size_t p = bpack_idx(rbase + v, col);          // store in B-pair layout
        yr0[p] =  accr0[v];
        yi0[p] = -acci0[v];
        yr1[p] =  accr1[v];
        yi1[p] = -acci1[v];
    }
}

// ---------------------------------------------------------------------------
// Kernel 3: column DFT (complex).  Per image:
//   Re = cos @ Yr + sin @ Yi,  Im = cos @ Yi - sin @ Yr
// A-side twiddle rows contiguous b64; B-side Yr/Yi pair-packed b64.
// Per k-step: 4 b64 loads (all immediate-offset) -> 4 WMMA.
// ---------------------------------------------------------------------------
__global__ void __launch_bounds__(256)
k_coldft(const float* __restrict__ twc, const float* __restrict__ tws,
         const float* __restrict__ Yrp, const float* __restrict__ Yip,
         float* __restrict__ Re, float* __restrict__ Im) {
    int wave = blockIdx.x * 8 + (threadIdx.x >> 5);    // 8192 waves total
    int lane = threadIdx.x & 31;
    int img = wave >> 9;
    int rem = wave & 511;
    int m0 = (rem >> 4) << 4;
    int n0 = (rem & 15) << 4;
    int ln    = lane & 15;
    int khalf = lane >> 4;
    int koff  = khalf << 1;
    int arow  = m0 + ln;

    const float* pac = twc + (size_t)arow * 512 + koff;
    const float* pas = tws + (size_t)arow * 512 + koff;
    size_t bp0 = ((size_t)((n0 + ln) * 2 + khalf)) << 1;
    const float* pbr = Yrp + (size_t)img * (512u * 256u) + bp0;
    const float* pbi = Yip + (size_t)img * (512u * 256u) + bp0;

    v8f accRe  = {};
    v8f accImC = {};
    v8f accImS = {};
#pragma unroll 8
    for (int kb = 0; kb < 128; kb++) {
        v2f ac = *(const v2f*)(pac + (size_t)kb * 4);
        v2f as = *(const v2f*)(pas + (size_t)kb * 4);
        v2f br = *(const v2f*)(pbr + (size_t)kb * 1024);
        v2f bi = *(const v2f*)(pbi + (size_t)kb * 1024);
        accRe  = WMMA_F32X4(ac, br, accRe);
        accRe  = WMMA_F32X4(as, bi, accRe);
        accImC = WMMA_F32X4(ac, bi, accImC);
        accImS = WMMA_F32X4(as, br, accImS);   // subtracted at store (f32 WMMA has no A/B neg)
    }
    float* re = Re + (size_t)img * (512u * 256u);
    float* im = Im + (size_t)img * (512u * 256u);
    int rbase = m0 + (khalf << 3);
    int col = n0 + ln;
#pragma unroll
    for (int v = 0; v < 8; v++) {
        re[(size_t)(rbase + v) * 256 + col] = accRe[v];
        im[(size_t)(rbase + v) * 256 + col] = accImC[v] - accImS[v];
    }
}

// ---------------------------------------------------------------------------
// Kernel 4: per-part 3x3 conv (pad=1) + bias, then cumprod over batch.
// Parts: p0 = Re rows 0..255 (low.re), p1 = Im rows 0..255 (low.im),
//        p2 = Re rows 256..511 (high.re), p3 = Im rows 256..511 (high.im).
// sdi_out layout: [p][b][i][j]
// ---------------------------------------------------------------------------
__global__ void k_sdi(const float* __restrict__ Re, const float* __restrict__ Im,
                      const float* __restrict__ w9, const float* __restrict__ bias,
                      float* __restrict__ sdi_out) {
    int idx = blockIdx.x * blockDim.x + threadIdx.x;   // 4*256*256
    int j = idx & 255;
    int i = (idx >> 8) & 255;
    int p = idx >> 16;
    const float* base = (p & 1) ? Im : Re;
    size_t rowoff = (p >> 1) ? (size_t)256 * 256 : 0;  // high half starts at row 256
    float w[9];
#pragma unroll
    for (int t = 0; t < 9; t++) w[t] = w9[t];
    float bsv = bias[0];
    float prod = 1.0f;
    float* outp = sdi_out + (size_t)p * (16u * 65536u) + (size_t)i * 256 + j;
    for (int b = 0; b < 16; b++) {
        const float* pb = base + (size_t)b * (512u * 256u) + rowoff;
        float acc = bsv;
#pragma unroll
        for (int di = 0; di < 3; di++) {
            int r = i + di - 1;
            if (r < 0 || r > 255) continue;
#pragma unroll
            for (int dj = 0; dj < 3; dj++) {
                int c = j + dj - 1;
                if (c < 0 || c > 255) continue;
                acc += pb[(size_t)r * 256 + c] * w[di * 3 + dj];
            }
        }
        prod *= acc;                                   // cumprod over batch
        outp[(size_t)b * 65536u] = prod;
    }
}

// ---------------------------------------------------------------------------
// Kernel 5: fuse 1x1 conv (4->1) + BN + SiLU, gen BN + ReLU, then broadcast
// multiply by w_final[64] into out (16,64,256,256).
// (softmax over singleton K*K axis == 1.0, so the gw branch is provably dead.)
// ---------------------------------------------------------------------------
__global__ void k_fuse_out(const float* __restrict__ sdi_out,
                           const float* __restrict__ fw, const float* __restrict__ fg,
                           const float* __restrict__ fb,
                           const float* __restrict__ gg, const float* __restrict__ gb,
                           const float* __restrict__ wf,
                           float* __restrict__ out) {
    int idx = blockIdx.x * blockDim.x + threadIdx.x;   // 16*65536
    int pix = idx & 65535;
    int b = idx >> 16;
    size_t o1 = (size_t)b * 65536u + pix;
    const size_t ps = (size_t)16 * 65536u;
    float z = sdi_out[o1]          * fw[0]
            + sdi_out[o1 + ps]     * fw[1]
            + sdi_out[o1 + 2 * ps] * fw[2]
            + sdi_out[o1 + 3 * ps] * fw[3];
    float sf = fg[0] * rsqrtf(1.0f + 1e-5f);
    z = z * sf + fb[0];
    z = z / (1.0f + __expf(-z));                       // SiLU
    float sg = gg[0] * rsqrtf(1.0f + 1e-5f);
    z = z * sg + gb[0];
    z = fmaxf(z, 0.0f);                                // ReLU
    float* ob = out + (size_t)b * (64u * 65536u) + pix;
#pragma unroll
    for (int o = 0; o < 64; o++)
        ob[(size_t)o * 65536u] = z * wf[o];            // coalesced per-plane stores
}

// ---------------------------------------------------------------------------
extern "C" void kernel_launch(void* const* d_in, const int* in_sizes, int n_in,
                              void* d_out, int out_size, void* d_ws, size_t ws_size,
                              hipStream_t stream) {
    (void)in_sizes; (void)n_in; (void)out_size; (void)ws_size;
    const float* x         = (const float*)d_in[0];
    // gw branch (d_in[1..8]) is dead: softmax over K*K==1 axis is identically 1
    const float* w_spd_gf  = (const float*)d_in[9];
    const float* bn_g_gf   = (const float*)d_in[10];
    const float* bn_b_gf   = (const float*)d_in[11];
    const float* sdi_w_gf  = (const float*)d_in[12];
    const float* sdi_b_gf  = (const float*)d_in[13];
    const float* fuse_w_gf = (const float*)d_in[14];
    const float* fuse_g_gf = (const float*)d_in[15];
    const float* fuse_b_gf = (const float*)d_in[16];
    const float* gen_g     = (const float*)d_in[17];
    const float* gen_b     = (const float*)d_in[18];
    const float* w_final   = (const float*)d_in[19];

    float* ws   = (float*)d_ws;
    float* twc  = ws;                   // 512*512              =  262144 f
    float* tws  = twc  + 262144;        // 512*512
    float* twpc = tws  + 262144;        // 512*256 (packed)     =  131072 f
    float* twps = twpc + 131072;
    float* ypad = twps + 131072;        // 16*512*512           = 4194304 f
    float* Yrp  = ypad + 4194304;       // 16*512*256 (packed)  = 2097152 f
    float* Yip  = Yrp  + 2097152;
    float* Re   = Yip  + 2097152;       // 16*512*256
    float* Im   = Re   + 2097152;
    float* sdio = Im   + 2097152;       // 4*16*256*256         = 4194304 f
                                        // total ~67 MB (fits L2)

    k_twiddle <<<262144 / 256, 256, 0, stream>>>(twc, tws, twpc, twps);
    k_spd_pad <<<(16 * 512 * 512) / 256, 256, 0, stream>>>(x, w_spd_gf, bn_g_gf, bn_b_gf, ypad);
    k_rowdft  <<<512, 256, 0, stream>>>(ypad, twpc, twps, Yrp, Yip);
    k_coldft  <<<1024, 256, 0, stream>>>(twc, tws, Yrp, Yip, Re, Im);
    k_sdi     <<<(4 * 256 * 256) / 256, 256, 0, stream>>>(Re, Im, sdi_w_gf, sdi_b_gf, sdio);
    k_fuse_out<<<(16 * 65536) / 256, 256, 0, stream>>>(sdio, fuse_w_gf, fuse_g_gf, fuse_b_gf,
                                                       gen_g, gen_b, w_final, (float*)d_out);
}